// ExportableGENConv_1649267441699
// MI455X (gfx1250) — compile-verified
//
#include <hip/hip_runtime.h>
#include <hip/hip_bf16.h>

typedef float v2f __attribute__((ext_vector_type(2)));
typedef float v4f __attribute__((ext_vector_type(4)));
typedef float v8f __attribute__((ext_vector_type(8)));

#define DI 64      // in/out channels
#define HH 128     // mlp hidden

// D = A(16x4) * B(4x16) + C, fp32 WMMA, wave32.
__device__ __forceinline__ v8f wmma_f32_k4(v2f a, v2f b, v8f c) {
    return __builtin_amdgcn_wmma_f32_16x16x4_f32(false, a, false, b, (short)0, c, false, false);
}

// ---------------------------------------------------------------------------
// Kernel 1: msg[e] = relu(x[src[e]] + edge_attr[e] @ W_edge) + 1e-7
// block = 128 (4 waves); each wave: 16 edges x 64 cols (4 WMMA tiles, K=64)
// edge_attr is streamed once -> NT loads; msg is a stream buffer -> NT stores.
// x gathers are reused across edges -> regular (RT) so they live in L2.
// ---------------------------------------------------------------------------
__global__ __launch_bounds__(128)
void k_edge_msg(const float* __restrict__ x, const int* __restrict__ src,
                const float* __restrict__ ea, const float* __restrict__ Wedge,
                float* __restrict__ msg, long long E, long long N) {
    __shared__ float Ws[64][66];      // W_edge, padded vs bank conflicts
    __shared__ float As[4][16][66];   // per-wave edge tiles

    const int tid  = threadIdx.x;
    const int wave = tid >> 5;
    const int lane = tid & 31;
    const int hf   = lane >> 4;       // half-wave select
    const int lm   = lane & 15;

    for (int i = tid; i < 64 * 64; i += 128)
        Ws[i >> 6][i & 63] = Wedge[i];

    const long long e0 = ((long long)blockIdx.x * 4 + wave) * 16;
    for (int r = 0; r < 16; ++r) {
        long long e  = e0 + r;
        long long ec = e < E ? e : E - 1;
        const v2f av = __builtin_nontemporal_load((const v2f*)(ea + ec * DI + lane * 2));
        As[wave][r][lane * 2 + 0] = av.x;
        As[wave][r][lane * 2 + 1] = av.y;
    }
    __syncthreads();

    for (int n0 = 0; n0 < DI; n0 += 16) {
        v8f acc = {};
#pragma unroll
        for (int k0 = 0; k0 < DI; k0 += 4) {
            const int kb = k0 + hf * 2;
            v2f a, b;
            a.x = As[wave][lm][kb];     a.y = As[wave][lm][kb + 1];
            b.x = Ws[kb][n0 + lm];      b.y = Ws[kb + 1][n0 + lm];
            acc = wmma_f32_k4(a, b, acc);
        }
#pragma unroll
        for (int i = 0; i < 8; ++i) {
            const int m = i + hf * 8;
            const long long e = e0 + m;
            if (e < E) {
                const long long s = (long long)src[e];
                float v = acc[i] + x[s * DI + n0 + lm];
                v = (v > 0.f ? v : 0.f) + 1e-7f;
                __builtin_nontemporal_store(v, msg + e * DI + n0 + lm);
            }
        }
    }
}

// ---------------------------------------------------------------------------
// Kernel 2: segment max (msg > 0, so uint bit-pattern atomicMax == float max)
// msg streamed (NT); accumulators stay RT -> resident in L2 for the atomics.
// ---------------------------------------------------------------------------
__global__ __launch_bounds__(256)
void k_seg_max(const float* __restrict__ msg, const int* __restrict__ dst,
               unsigned int* __restrict__ mx, long long E) {
    const long long i = (long long)blockIdx.x * blockDim.x + threadIdx.x;
    if (i >= E * 16) return;
    const long long e = i >> 4;
    const int d4 = (int)(i & 15) * 4;
    const long long dd = (long long)dst[e];
    const v4f m = __builtin_nontemporal_load((const v4f*)(msg + e * DI + d4));
    unsigned int* b = mx + dd * DI + d4;
    atomicMax(b + 0, __float_as_uint(m.x));
    atomicMax(b + 1, __float_as_uint(m.y));
    atomicMax(b + 2, __float_as_uint(m.z));
    atomicMax(b + 3, __float_as_uint(m.w));
}

// ---------------------------------------------------------------------------
// Kernel 3: sum_per[dst] += exp(msg - max[dst])
// ---------------------------------------------------------------------------
__global__ __launch_bounds__(256)
void k_seg_expsum(const float* __restrict__ msg, const int* __restrict__ dst,
                  const float* __restrict__ mx, float* __restrict__ sum, long long E) {
    const long long i = (long long)blockIdx.x * blockDim.x + threadIdx.x;
    if (i >= E * 16) return;
    const long long e = i >> 4;
    const int d4 = (int)(i & 15) * 4;
    const long long dd = (long long)dst[e];
    const v4f m  = __builtin_nontemporal_load((const v4f*)(msg + e * DI + d4));
    const v4f mv = *(const v4f*)(mx + dd * DI + d4);
    float* b = sum + dd * DI + d4;
    atomicAdd(b + 0, __expf(m.x - mv.x));
    atomicAdd(b + 1, __expf(m.y - mv.y));
    atomicAdd(b + 2, __expf(m.z - mv.z));
    atomicAdd(b + 3, __expf(m.w - mv.w));
}

// ---------------------------------------------------------------------------
// Kernel 4: agg[dst] += msg * exp(msg - max[dst]) / (sum[dst] + 1e-16)
// ---------------------------------------------------------------------------
__global__ __launch_bounds__(256)
void k_seg_agg(const float* __restrict__ msg, const int* __restrict__ dst,
               const float* __restrict__ mx, const float* __restrict__ sum,
               float* __restrict__ agg, long long E) {
    const long long i = (long long)blockIdx.x * blockDim.x + threadIdx.x;
    if (i >= E * 16) return;
    const long long e = i >> 4;
    const int d4 = (int)(i & 15) * 4;
    const long long dd = (long long)dst[e];
    const v4f m  = __builtin_nontemporal_load((const v4f*)(msg + e * DI + d4));
    const v4f mv = *(const v4f*)(mx + dd * DI + d4);
    const v4f sv = *(const v4f*)(sum + dd * DI + d4);
    float* b = agg + dd * DI + d4;
    atomicAdd(b + 0, m.x * __expf(m.x - mv.x) / (sv.x + 1e-16f));
    atomicAdd(b + 1, m.y * __expf(m.y - mv.y) / (sv.y + 1e-16f));
    atomicAdd(b + 2, m.z * __expf(m.z - mv.z) / (sv.z + 1e-16f));
    atomicAdd(b + 3, m.w * __expf(m.w - mv.w) / (sv.w + 1e-16f));
}

// ---------------------------------------------------------------------------
// Kernel 5: h = (agg + x) @ W1  (N x 128), plus per-column sum / sum-sq
// block = 128 (4 waves); wave strip = 16 rows x 128 cols (8 tiles, K=64)
// h is re-read by k_out and fits L2 (25.6MB) -> keep RT stores.
// ---------------------------------------------------------------------------
__global__ __launch_bounds__(128)
void k_h_stats(const float* __restrict__ x, const float* __restrict__ agg,
               const float* __restrict__ W1, float* __restrict__ h,
               float* __restrict__ colsum, float* __restrict__ colsq, long long N) {
    __shared__ float W1s[64][130];
    __shared__ float As[4][16][66];
    __shared__ float cs_s[128], cq_s[128];

    const int tid  = threadIdx.x;
    const int wave = tid >> 5;
    const int lane = tid & 31;
    const int hf   = lane >> 4;
    const int lm   = lane & 15;

    for (int i = tid; i < 64 * HH; i += 128)
        W1s[i >> 7][i & 127] = W1[i];
    if (tid < 128) { cs_s[tid] = 0.f; cq_s[tid] = 0.f; }

    const long long r0 = ((long long)blockIdx.x * 4 + wave) * 16;
    for (int r = 0; r < 16; ++r) {
        long long rr = r0 + r;
        long long rc = rr < N ? rr : N - 1;
        const float2 a  = *(const float2*)(agg + rc * DI + lane * 2);
        const float2 xv = *(const float2*)(x + rc * DI + lane * 2);
        As[wave][r][lane * 2 + 0] = a.x + xv.x;
        As[wave][r][lane * 2 + 1] = a.y + xv.y;
    }
    __syncthreads();

    for (int n0 = 0; n0 < HH; n0 += 16) {
        v8f acc = {};
#pragma unroll
        for (int k0 = 0; k0 < DI; k0 += 4) {
            const int kb = k0 + hf * 2;
            v2f a, b;
            a.x = As[wave][lm][kb];   a.y = As[wave][lm][kb + 1];
            b.x = W1s[kb][n0 + lm];   b.y = W1s[kb + 1][n0 + lm];
            acc = wmma_f32_k4(a, b, acc);
        }
        float p1 = 0.f, p2 = 0.f;
#pragma unroll
        for (int i = 0; i < 8; ++i) {
            const int m = i + hf * 8;
            const long long rr = r0 + m;
            if (rr < N) {
                const float v = acc[i];
                h[rr * HH + n0 + lm] = v;
                p1 += v;
                p2 += v * v;
            }
        }
        p1 += __shfl_xor(p1, 16, 32);   // combine the two half-wave rows of same col
        p2 += __shfl_xor(p2, 16, 32);
        if (hf == 0) {
            atomicAdd(&cs_s[n0 + lm], p1);
            atomicAdd(&cq_s[n0 + lm], p2);
        }
    }
    __syncthreads();
    if (tid < 128) {
        atomicAdd(colsum + tid, cs_s[tid]);
        atomicAdd(colsq + tid, cq_s[tid]);
    }
}

// ---------------------------------------------------------------------------
// Kernel 6: fold BN stats into per-column scale/shift (one block, 128 threads)
// ---------------------------------------------------------------------------
__global__ __launch_bounds__(128)
void k_bn(const float* __restrict__ colsum, const float* __restrict__ colsq,
          const float* __restrict__ gamma, const float* __restrict__ beta,
          float* __restrict__ scale, float* __restrict__ shift, float invN) {
    const int i = threadIdx.x;
    const float mu  = colsum[i] * invN;
    const float var = colsq[i] * invN - mu * mu;
    const float sc  = rsqrtf(var + 1e-5f) * gamma[i];
    scale[i] = sc;
    shift[i] = beta[i] - mu * sc;
}

// ---------------------------------------------------------------------------
// Kernel 7: out = relu(h*scale + shift) @ W2  (N x 64)
// block = 64 (2 waves, keeps static LDS < 64KB); wave strip = 16 rows, K=128
// h read should hit L2 (RT); final out is write-once -> NT stores.
// ---------------------------------------------------------------------------
__global__ __launch_bounds__(64)
void k_out(const float* __restrict__ h, const float* __restrict__ W2,
           const float* __restrict__ scale, const float* __restrict__ shift,
           float* __restrict__ out, long long N) {
    __shared__ float W2s[HH][66];
    __shared__ float As[2][16][130];
    __shared__ float sc_s[128], sh_s[128];

    const int tid  = threadIdx.x;
    const int wave = tid >> 5;
    const int lane = tid & 31;
    const int hf   = lane >> 4;
    const int lm   = lane & 15;

    for (int i = tid; i < HH * DI; i += 64)
        W2s[i >> 6][i & 63] = W2[i];
    for (int i = tid; i < 128; i += 64) { sc_s[i] = scale[i]; sh_s[i] = shift[i]; }
    __syncthreads();  // sc_s/sh_s needed for A staging below

    const long long r0 = ((long long)blockIdx.x * 2 + wave) * 16;
    for (int r = 0; r < 16; ++r) {
        long long rr = r0 + r;
        long long rc = rr < N ? rr : N - 1;
        const int c = lane * 4;
        const v4f hv = *(const v4f*)(h + rc * HH + c);
        float v0 = hv.x * sc_s[c + 0] + sh_s[c + 0];
        float v1 = hv.y * sc_s[c + 1] + sh_s[c + 1];
        float v2 = hv.z * sc_s[c + 2] + sh_s[c + 2];
        float v3 = hv.w * sc_s[c + 3] + sh_s[c + 3];
        As[wave][r][c + 0] = v0 > 0.f ? v0 : 0.f;
        As[wave][r][c + 1] = v1 > 0.f ? v1 : 0.f;
        As[wave][r][c + 2] = v2 > 0.f ? v2 : 0.f;
        As[wave][r][c + 3] = v3 > 0.f ? v3 : 0.f;
    }
    __syncthreads();

    for (int n0 = 0; n0 < DI; n0 += 16) {
        v8f acc = {};
#pragma unroll
        for (int k0 = 0; k0 < HH; k0 += 4) {
            const int kb = k0 + hf * 2;
            v2f a, b;
            a.x = As[wave][lm][kb];   a.y = As[wave][lm][kb + 1];
            b.x = W2s[kb][n0 + lm];   b.y = W2s[kb + 1][n0 + lm];
            acc = wmma_f32_k4(a, b, acc);
        }
#pragma unroll
        for (int i = 0; i < 8; ++i) {
            const int m = i + hf * 8;
            const long long rr = r0 + m;
            if (rr < N) __builtin_nontemporal_store(acc[i], out + rr * DI + n0 + lm);
        }
    }
}

// ---------------------------------------------------------------------------
extern "C" void kernel_launch(void* const* d_in, const int* in_sizes, int n_in,
                              void* d_out, int out_size, void* d_ws, size_t ws_size,
                              hipStream_t stream) {
    const float* x     = (const float*)d_in[0];
    const int*   eidx  = (const int*)d_in[1];
    const float* ea    = (const float*)d_in[2];
    const float* Wedge = (const float*)d_in[3];
    const float* W1    = (const float*)d_in[4];
    const float* gamma = (const float*)d_in[5];
    const float* beta  = (const float*)d_in[6];
    const float* W2    = (const float*)d_in[7];
    float* out = (float*)d_out;

    const long long N = in_sizes[0] / DI;
    const long long E = in_sizes[1] / 2;
    const int* src = eidx;
    const int* dst = eidx + E;

    // workspace layout (floats): msg | mx | sum | agg | colsum | colsq | scale | shift | h
    float* ws   = (float*)d_ws;
    float* msg  = ws;
    float* mx   = msg + (size_t)E * DI;
    float* sum  = mx  + (size_t)N * DI;
    float* agg  = sum + (size_t)N * DI;
    float* cs   = agg + (size_t)N * DI;
    float* cq   = cs + 128;
    float* sc   = cq + 128;
    float* sh   = sc + 128;
    float* hbuf = sh + 128;

    // zero the accumulators (mx..colsq) — stream-ordered, capture-safe
    hipMemsetAsync(mx, 0, ((size_t)3 * N * DI + 256) * sizeof(float), stream);

    k_edge_msg<<<(int)((E + 63) / 64), 128, 0, stream>>>(x, src, ea, Wedge, msg, E, N);

    const long long tot = E * 16;
    const int segBlocks = (int)((tot + 255) / 256);
    k_seg_max<<<segBlocks, 256, 0, stream>>>(msg, dst, (unsigned int*)mx, E);
    k_seg_expsum<<<segBlocks, 256, 0, stream>>>(msg, dst, mx, sum, E);
    k_seg_agg<<<segBlocks, 256, 0, stream>>>(msg, dst, mx, sum, agg, E);

    const int strips = (int)((N + 15) / 16);
    k_h_stats<<<(strips + 3) / 4, 128, 0, stream>>>(x, agg, W1, hbuf, cs, cq, N);
    k_bn<<<1, 128, 0, stream>>>(cs, cq, gamma, beta, sc, sh, 1.0f / (float)N);
    k_out<<<(strips + 1) / 2, 64, 0, stream>>>(hbuf, W2, sc, sh, out, N);
}